// DotGatConv_44263932952985
// MI455X (gfx1250) — compile-verified
//
#include <hip/hip_runtime.h>
#include <hip/hip_bf16.h>
#include <math.h>

#define D_IN 128
#define NH   4      // heads
#define NF   16     // features per head
#define HF   64     // NH*NF

typedef __attribute__((ext_vector_type(2))) float v2f;
typedef __attribute__((ext_vector_type(8))) float v8f;

// ---------------------------------------------------------------------------
// Init: segmax = -inf, denom = 0, out = 0 (grid-stride; deterministic each call)
// ---------------------------------------------------------------------------
__global__ void dgat_init_kernel(float* __restrict__ segmax,
                                 float* __restrict__ denom,
                                 float* __restrict__ out,
                                 int nh, int nhf) {
    int i = blockIdx.x * blockDim.x + threadIdx.x;
    int stride = gridDim.x * blockDim.x;
    for (int j = i; j < nh; j += stride) {
        segmax[j] = -__builtin_huge_valf();
        denom[j]  = 0.0f;
    }
    for (int j = i; j < nhf; j += stride) {
        out[j] = 0.0f;
    }
}

// ---------------------------------------------------------------------------
// Projection ft = feat @ W using V_WMMA_F32_16X16X4_F32 (exact f32 path).
// One wave computes one 16x16 tile of ft; K=128 -> 32 WMMA ops per tile.
//
// A (16x4 f32) layout: lanes 0-15 hold (K=0,K=1), lanes 16-31 hold (K=2,K=3),
// row M = lane&15. B (4x16) mirrors with N = lane&15. C/D: VGPR v holds
// row v + 8*(lane>=16), col lane&15.
// ---------------------------------------------------------------------------
__global__ void dgat_proj_wmma_kernel(const float* __restrict__ feat,
                                      const float* __restrict__ W,
                                      float* __restrict__ ft,
                                      int N, int ntiles) {
    const int lane = threadIdx.x & 31;
    const int wave = threadIdx.x >> 5;
    const int tile = blockIdx.x * (blockDim.x >> 5) + wave;
    if (tile >= ntiles) return;              // wave-uniform: EXEC stays all-ones

    const int mt = tile >> 2;                // 16-row tile index
    const int nt = tile & 3;                 // 16-col tile index (HF=64 -> 4 tiles)
    const int m0 = mt * 16;
    const int n0 = nt * 16;

    const int half = lane >> 4;              // 0: K={0,1}, 1: K={2,3}
    const int l    = lane & 15;

    int rowA = m0 + l;                       // clamp (no-op when N % 16 == 0)
    if (rowA > N - 1) rowA = N - 1;
    const float* __restrict__ arow = feat + (size_t)rowA * D_IN;
    const float* __restrict__ bp   = W + n0 + l;

    v8f c = {0.f, 0.f, 0.f, 0.f, 0.f, 0.f, 0.f, 0.f};

#pragma unroll 8
    for (int k = 0; k < D_IN; k += 4) {
        const int kk = k + 2 * half;
        v2f a, b;
        a.x = arow[kk];                      // merged to a single b64 load
        a.y = arow[kk + 1];
        b.x = bp[(size_t)kk * HF];
        b.y = bp[(size_t)(kk + 1) * HF];
        // 8 args: (neg_a, A, neg_b, B, c_mod, C, reuse_a, reuse_b)
        c = __builtin_amdgcn_wmma_f32_16x16x4_f32(
                false, a, false, b, (short)0, c, false, false);
    }

    // Tile-level (wave-uniform) bounds check: full tiles take the unguarded
    // straight-line store path; only a ragged tail tile predicates per row.
    float* __restrict__ op = ft + ((size_t)(m0 + 8 * half)) * HF + n0 + l;
    if (m0 + 16 <= N) {
#pragma unroll
        for (int v = 0; v < 8; ++v)
            op[(size_t)v * HF] = c[v];
    } else {
#pragma unroll
        for (int v = 0; v < 8; ++v) {
            const int row = m0 + v + 8 * half;
            if (row < N)
                ft[(size_t)row * HF + n0 + l] = c[v];
        }
    }
}

// ---------------------------------------------------------------------------
// Exact float atomic max via signed/unsigned int punning (IEEE order-preserving)
// ---------------------------------------------------------------------------
__device__ __forceinline__ void atomicMaxFloat(float* addr, float val) {
    if (val >= 0.0f)
        atomicMax((int*)addr, __float_as_int(val));
    else
        atomicMin((unsigned int*)addr, __float_as_uint(val));
}

// ---------------------------------------------------------------------------
// Per-(edge,head) scaled dot score + segment max.  t = edge*4 + h, so the
// 4 threads of a quad read consecutive 64B chunks of the same ft row.
// ---------------------------------------------------------------------------
__global__ void dgat_score_kernel(const float* __restrict__ ft,
                                  const int* __restrict__ src,
                                  const int* __restrict__ dst,
                                  float* __restrict__ e,
                                  float* __restrict__ segmax,
                                  int E) {
    const int t = blockIdx.x * blockDim.x + threadIdx.x;
    if (t >= E * NH) return;
    const int edge = t >> 2;
    const int h    = t & 3;
    const int s    = src[edge];
    const int d    = dst[edge];

    const float4* __restrict__ a = (const float4*)(ft + (size_t)s * HF + h * NF);
    const float4* __restrict__ b = (const float4*)(ft + (size_t)d * HF + h * NF);
    float acc = 0.0f;
#pragma unroll
    for (int i = 0; i < 4; ++i) {
        const float4 av = a[i], bv = b[i];
        acc += av.x * bv.x + av.y * bv.y + av.z * bv.z + av.w * bv.w;
    }
    acc *= 0.25f;                            // 1/sqrt(F), F=16
    e[t] = acc;
    atomicMaxFloat(&segmax[(size_t)d * NH + h], acc);
}

// ---------------------------------------------------------------------------
// ex = exp(e - segmax[dst]);  denom[dst] += ex   (e overwritten in place)
// ---------------------------------------------------------------------------
__global__ void dgat_exp_kernel(const float* __restrict__ segmax,
                                const int* __restrict__ dst,
                                float* __restrict__ e,
                                float* __restrict__ denom,
                                int E) {
    const int t = blockIdx.x * blockDim.x + threadIdx.x;
    if (t >= E * NH) return;
    const int edge = t >> 2;
    const int h    = t & 3;
    const int d    = dst[edge];
    const float ex = __expf(e[t] - segmax[(size_t)d * NH + h]);
    e[t] = ex;
    atomicAdd(&denom[(size_t)d * NH + h], ex);
}

// ---------------------------------------------------------------------------
// out[dst,h,:] += ft[src,h,:] * (ex / denom[dst,h])  via f32 hardware atomics
// ---------------------------------------------------------------------------
__global__ void dgat_agg_kernel(const float* __restrict__ ft,
                                const int* __restrict__ src,
                                const int* __restrict__ dst,
                                const float* __restrict__ ex,
                                const float* __restrict__ denom,
                                float* __restrict__ out,
                                int E) {
    const int t = blockIdx.x * blockDim.x + threadIdx.x;
    if (t >= E * NH) return;
    const int edge = t >> 2;
    const int h    = t & 3;
    const int s    = src[edge];
    const int d    = dst[edge];
    const float sa = ex[t] / denom[(size_t)d * NH + h];

    const float* __restrict__ fs = ft + (size_t)s * HF + h * NF;
    float* __restrict__ op = out + (size_t)d * HF + h * NF;
#pragma unroll
    for (int i = 0; i < NF; ++i)
        atomicAdd(&op[i], fs[i] * sa);
}

// ---------------------------------------------------------------------------
// Launch
// ---------------------------------------------------------------------------
extern "C" void kernel_launch(void* const* d_in, const int* in_sizes, int n_in,
                              void* d_out, int out_size, void* d_ws, size_t ws_size,
                              hipStream_t stream) {
    const float* feat = (const float*)d_in[0];   // [N, 128]
    const float* W    = (const float*)d_in[1];   // [128, 64]
    const int*   src  = (const int*)d_in[2];     // [E]
    const int*   dst  = (const int*)d_in[3];     // [E]
    float*       out  = (float*)d_out;           // [N, 4, 16]

    const int N = in_sizes[0] / D_IN;
    const int E = in_sizes[2];

    // Workspace layout (floats): ft[N*64] | e[E*4] | segmax[N*4] | denom[N*4]
    float* ws     = (float*)d_ws;
    float* ft     = ws;
    float* e      = ft + (size_t)N * HF;
    float* segmax = e + (size_t)E * NH;
    float* denom  = segmax + (size_t)N * NH;

    const int nh  = N * NH;
    const int nhf = N * HF;

    dgat_init_kernel<<<2048, 256, 0, stream>>>(segmax, denom, out, nh, nhf);

    const int mtiles = (N + 15) / 16;
    const int ntiles = mtiles * 4;               // HF/16 = 4 col tiles
    dgat_proj_wmma_kernel<<<(ntiles + 7) / 8, 256, 0, stream>>>(feat, W, ft, N, ntiles);

    const int work = E * NH;
    const int blk  = 256;
    const int grd  = (work + blk - 1) / blk;
    dgat_score_kernel<<<grd, blk, 0, stream>>>(ft, src, dst, e, segmax, E);
    dgat_exp_kernel<<<grd, blk, 0, stream>>>(segmax, dst, e, denom, E);
    dgat_agg_kernel<<<grd, blk, 0, stream>>>(ft, src, dst, e, denom, out, E);
}